// My_parse_cnn_layer_88167088652529
// MI455X (gfx1250) — compile-verified
//
#include <hip/hip_runtime.h>
#include <cstdint>
#include <cstddef>

namespace {

constexpr int kB = 64, kW = 256, kH = 256, kU = 9;
constexpr int TI = 16, TJ = 64, TILE = TI * TJ;   // 1024 outputs per spatial tile
constexpr int HROWS = TI + 2;                     // 18 staged input rows (halo)
constexpr int LROW  = 68;                         // staged row stride in floats (17 x float4)
constexpr int NCHUNK = HROWS * 17;                // 306 b128 chunks per input stage
constexpr int NT = 256;                           // 8 wave32 waves
constexpr int NBATCH = 8;                         // batches handled per block

// LDS layout (in floats)
constexpr int W_OFF  = 0;                         // 9216: weight tile, [pos*9+k]
constexpr int BS_OFF = TILE * kU;                 // 1024: per-position bias sums
constexpr int IN_OFF = BS_OFF + TILE;             // 2 x 1224: input double buffer
constexpr int IN_BUF = HROWS * LROW;              // 1224
constexpr int SMEM_FLOATS = IN_OFF + 2 * IN_BUF;  // 12688 floats = 50752 B

__device__ __forceinline__ uint32_t lds_addr(const void* p) {
  // flat address of an LDS object: low 32 bits are the wave-relative LDS offset
  return (uint32_t)(uintptr_t)p;
}

__device__ __forceinline__ void async_ld_b128(uint32_t lds, const float* g) {
  // CDNA5 async global->LDS DMA, tracked with ASYNCcnt
  asm volatile("global_load_async_to_lds_b128 %0, %1, off"
               :: "v"(lds), "v"(g) : "memory");
}
__device__ __forceinline__ void wait_async_0() {
  asm volatile("s_wait_asynccnt 0x0" ::: "memory");
}
__device__ __forceinline__ void wait_async_2() {
  asm volatile("s_wait_asynccnt 0x2" ::: "memory");
}

// Stage one (batch) input halo tile: 18 rows x 68 cols, 16B-aligned b128 chunks.
// Every thread issues exactly 2 async ops (tail clamped-duplicated) so EXEC is
// full and per-wave ASYNCcnt advances uniformly by 2 per stage.
__device__ __forceinline__ void stage_input(const float* __restrict__ in, int batch,
                                            int i0, int j0, float* buf, int tid) {
#pragma unroll
  for (int u = 0; u < 2; ++u) {
    int q = tid + u * NT;
    if (q > NCHUNK - 1) q = NCHUNK - 1;           // duplicate last chunk (coalesces in-wave)
    int row = q / 17, cq = q - row * 17;
    int gr = i0 + row;      if (gr > kW - 1) gr = kW - 1;   // clamped src: garbage x 0-weight
    int gc = j0 + cq * 4;   if (gc > kH - 4) gc = kH - 4;
    const float* ga = in + (size_t)batch * (kW * kH) + (size_t)gr * kH + gc;
    async_ld_b128(lds_addr(&buf[row * LROW + cq * 4]), ga);
  }
}

__global__ __launch_bounds__(NT)
void lconv_kernel(const float* __restrict__ in, const float* __restrict__ w,
                  const float* __restrict__ b, float* __restrict__ out) {
  extern __shared__ float smem[];
  float* s_w  = smem + W_OFF;
  float* s_bs = smem + BS_OFF;
  float* s_in = smem + IN_OFF;

  const int tid = threadIdx.x;
  const int j0  = blockIdx.x * TJ;
  const int i0  = blockIdx.y * TI;
  const int b0  = blockIdx.z * NBATCH;

  // ---- Phase A: async-stage bias tile into s_w (used as scratch) ----
  // 16 rows x 576 contiguous floats; every run is 16B aligned -> 2304 b128 = 9/thread.
#pragma unroll
  for (int u = 0; u < 9; ++u) {
    int chunk = tid + u * NT;
    int r = chunk / 144, cq = chunk - r * 144;
    const float* ga = b + ((size_t)(i0 + r) * kH + j0) * kU + cq * 4;
    async_ld_b128(lds_addr(&s_w[r * (TJ * kU) + cq * 4]), ga);
  }
  wait_async_0();
  __syncthreads();

  // ---- Phase B: per-position bias sums into registers ----
  float bs[TILE / NT];
#pragma unroll
  for (int p = 0; p < TILE / NT; ++p) {
    const float* bp = &s_w[(tid + p * NT) * kU];
    float s = 0.f;
#pragma unroll
    for (int k = 0; k < kU; ++k) s += bp[k];
    bs[p] = s;
  }
  __syncthreads();  // all bias reads done before weights overwrite s_w

  // ---- Phase C: async-stage weight tile + first input tile; park bias sums ----
#pragma unroll
  for (int u = 0; u < 9; ++u) {
    int chunk = tid + u * NT;
    int r = chunk / 144, cq = chunk - r * 144;
    const float* ga = w + ((size_t)(i0 + r) * kH + j0) * kU + cq * 4;
    async_ld_b128(lds_addr(&s_w[r * (TJ * kU) + cq * 4]), ga);
  }
  stage_input(in, b0, i0, j0, s_in, tid);         // batch 0 -> buffer 0
#pragma unroll
  for (int p = 0; p < TILE / NT; ++p) s_bs[tid + p * NT] = bs[p];
  wait_async_0();
  __syncthreads();  // all waves' weight DMA landed before anyone fixes up

  // Zero-weight trick: reference zero-pads the high side; zero the weights of
  // out-of-range taps instead of zeroing inputs. Only boundary tiles need it.
  if (i0 + TI + 1 >= kW || j0 + TJ + 1 >= kH) {
#pragma unroll
    for (int t = 0; t < (TILE * kU) / NT; ++t) {
      int idx = tid + t * NT;                     // < 9216
      int pos = idx / kU, k = idx - pos * kU;
      int r = pos >> 6, c = pos & 63;
      int m = k / 3, n = k - m * 3;               // k = m*3+n matches reference patch order
      if (i0 + r + m >= kW || j0 + c + n >= kH) s_w[idx] = 0.f;
    }
  }
  __syncthreads();

  // ---- Phase D: batch loop, double-buffered async input staging ----
  const int tq = tid >> 6;   // row group 0..3 -> rows 4*tq .. 4*tq+3
  const int c  = tid & 63;

  for (int bb = 0; bb < NBATCH; ++bb) {
    float* cur = s_in + (bb & 1) * IN_BUF;
    if (bb + 1 < NBATCH) {
      stage_input(in, b0 + bb + 1, i0, j0, s_in + ((bb + 1) & 1) * IN_BUF, tid);
      wait_async_2();        // oldest 2 (current buffer) done; next stage in flight
    } else {
      wait_async_0();
    }
    __syncthreads();

    // 6x3 register window shared across this thread's 4 vertically-adjacent outputs
    float xin[6][3];
#pragma unroll
    for (int rr = 0; rr < 6; ++rr) {
#pragma unroll
      for (int nn = 0; nn < 3; ++nn)
        xin[rr][nn] = cur[(tq * 4 + rr) * LROW + c + nn];
    }

    const int batch = b0 + bb;
#pragma unroll
    for (int dr = 0; dr < 4; ++dr) {
      const int r   = tq * 4 + dr;
      const int pos = (r << 6) | c;
      const float* wp = &s_w[pos * kU];
      float acc = 0.f;
#pragma unroll
      for (int m = 0; m < 3; ++m) {
#pragma unroll
        for (int n = 0; n < 3; ++n)
          acc = fmaf(xin[dr + m][n], wp[m * 3 + n], acc);
      }
      // Write-once output stream: non-temporal store keeps L2 for the reused
      // input halos / weight tiles (lowers to global_store th:TH_STORE_NT).
      float res = 9.0f * acc + s_bs[pos];
      float* op = out + (size_t)batch * (kW * kH) + (size_t)(i0 + r) * kH + (j0 + c);
      __builtin_nontemporal_store(res, op);
    }
    __syncthreads();  // protect cur buffer before it is re-staged next iteration
  }
}

}  // namespace

extern "C" void kernel_launch(void* const* d_in, const int* in_sizes, int n_in,
                              void* d_out, int out_size, void* d_ws, size_t ws_size,
                              hipStream_t stream) {
  (void)in_sizes; (void)n_in; (void)d_ws; (void)ws_size; (void)out_size;
  const float* in = (const float*)d_in[0];
  const float* w  = (const float*)d_in[1];
  const float* b  = (const float*)d_in[2];
  float* out = (float*)d_out;

  dim3 grid(kH / TJ, kW / TI, kB / NBATCH);  // (4, 16, 8) = 512 blocks
  dim3 block(NT);
  size_t shmem = SMEM_FLOATS * sizeof(float);
  lconv_kernel<<<grid, block, shmem, stream>>>(in, w, b, out);
}